// Capsule_fc_6373731467780
// MI455X (gfx1250) — compile-verified
//
#include <hip/hip_runtime.h>

// Capsule routing, fused: u_hat (189 MB) is never materialized.
//   s[b,o,d] = sum_{i,k} (c[i,o]*W[i,o,d,k]) * x[b,i,k]   -> f32 WMMA GEMM
//   b[i,o] += sum_{b,d} u_hat[b,i,o,d]*s[b,o,d]           -> LDS-staged FMA kernel
// x : [256,1152,8] f32, W : [1152,10,16,8] f32, out : [256,10,16] f32

typedef __attribute__((ext_vector_type(2))) float v2f;
typedef __attribute__((ext_vector_type(8))) float v8f;

#define BATCH  256
#define N_IN   1152
#define N_OUT  10
#define D_IN   8
#define D_OUT  16
#define KS     4     // K-split across blocks
#define WAVES  4     // waves per GEMM block (K-split within block)

// CDNA5 async global->LDS staging, if the toolchain declares the builtins.
#if defined(__AMDGCN__) && \
    __has_builtin(__builtin_amdgcn_global_load_async_to_lds_b32) && \
    __has_builtin(__builtin_amdgcn_s_wait_asynccnt)
#define USE_ASYNC_LDS 1
typedef __attribute__((address_space(1))) int g1int;   // global (device) AS
typedef __attribute__((address_space(3))) int l3int;   // LDS (shared) AS
#else
#define USE_ASYNC_LDS 0
#endif

// ---------------------------------------------------------------- zero init
__global__ void zero_kernel(float* __restrict__ p, int n) {
    int i = blockIdx.x * blockDim.x + threadIdx.x;
    int stride = gridDim.x * blockDim.x;
    for (; i < n; i += stride) p[i] = 0.0f;
}

// ------------------------------------------- softmax over o + zero s_raw
// launch <<<9,128>>> : exactly 1152 threads, one per input capsule row
__global__ void softmax_zero_kernel(const float* __restrict__ blog,
                                    float* __restrict__ c,
                                    float* __restrict__ s_raw) {
    const int tid = blockIdx.x * blockDim.x + threadIdx.x;    // 0..1151
    const int nthr = gridDim.x * blockDim.x;
    // zero the s accumulator for this routing iteration
    for (int e = tid; e < BATCH * N_OUT * D_OUT; e += nthr) s_raw[e] = 0.0f;

    if (tid < N_IN) {
        float v[N_OUT];
        float m = -1e30f;
#pragma unroll
        for (int o = 0; o < N_OUT; ++o) {
            v[o] = blog[tid * N_OUT + o];
            m = fmaxf(m, v[o]);
        }
        float sum = 0.0f;
#pragma unroll
        for (int o = 0; o < N_OUT; ++o) {
            v[o] = __expf(v[o] - m);
            sum += v[o];
        }
        const float inv = 1.0f / sum;
#pragma unroll
        for (int o = 0; o < N_OUT; ++o) c[tid * N_OUT + o] = v[o] * inv;
    }
}

// ------------------------------------------------------- fused s GEMM (WMMA)
// grid (16 m-tiles, 10 o, KS), block = 4 waves. Each wave owns a 16x16 f32
// accumulator tile (batch-tile x D_out) and marches K = (i,k) with
// V_WMMA_F32_16X16X4_F32. c[i,o] is uniform -> scalar load.
__global__ void __launch_bounds__(WAVES * 32)
s_gemm_kernel(const float* __restrict__ x, const float* __restrict__ W,
              const float* __restrict__ c, float* __restrict__ s_raw) {
    const int mt   = blockIdx.x;            // batch tile 0..15
    const int o    = blockIdx.y;            // 0..9
    const int ks   = blockIdx.z;            // 0..KS-1
    const int lane = threadIdx.x & 31;
    const int wave = threadIdx.x >> 5;
    const int kw   = ks * WAVES + wave;     // global K-worker 0..15

    const int row = mt * 16 + (lane & 15);  // batch row this lane feeds (A)
    const int kb  = (lane >> 4) * 2;        // A/B frag K sub-offset {0,2}
    const int d   = lane & 15;              // B column (D_out) this lane feeds

    const float* __restrict__ xrow = x + (size_t)row * (N_IN * D_IN);

    v8f acc;
#pragma unroll
    for (int j = 0; j < 8; ++j) acc[j] = 0.0f;

    // 1152 input capsules / 16 K-workers = 72 per wave, 2 WMMAs each
    for (int t = 0; t < N_IN / 16; ++t) {
        const int i = kw + t * 16;

        const float* __restrict__ xp = xrow + (size_t)i * D_IN + kb;
        const float* __restrict__ wp =
            W + (((size_t)i * N_OUT + o) * D_OUT + d) * D_IN + kb;

        // prefetch next K-step (L2-resident streams; emits global_prefetch_b8)
        __builtin_prefetch(xp + 16 * D_IN, 0, 1);
        __builtin_prefetch(wp + 16 * N_OUT * D_OUT * D_IN, 0, 1);

        v2f a0 = *(const v2f*)(xp);          // k = kb, kb+1
        v2f a1 = *(const v2f*)(xp + 4);      // k = kb+4, kb+5
        v2f w0 = *(const v2f*)(wp);
        v2f w1 = *(const v2f*)(wp + 4);

        const float ci = c[i * N_OUT + o];   // uniform -> s_load

        v2f b0; b0.x = ci * w0.x; b0.y = ci * w0.y;
        v2f b1; b1.x = ci * w1.x; b1.y = ci * w1.y;

        acc = __builtin_amdgcn_wmma_f32_16x16x4_f32(
            false, a0, false, b0, (short)0, acc, false, false);
        acc = __builtin_amdgcn_wmma_f32_16x16x4_f32(
            false, a1, false, b1, (short)0, acc, false, false);
    }

    // reduce the 4 per-wave tiles in LDS, then atomically fold the block's
    // partial tile into s_raw (KS blocks contribute per tile).
    __shared__ float red[WAVES][256];
#pragma unroll
    for (int j = 0; j < 8; ++j) red[wave][lane * 8 + j] = acc[j];
    __syncthreads();

    for (int e = threadIdx.x; e < 256; e += WAVES * 32) {
        float sum = red[0][e] + red[1][e] + red[2][e] + red[3][e];
        const int l  = e >> 3;                 // source lane
        const int j  = e & 7;                  // source vgpr
        const int M  = j + ((l >= 16) ? 8 : 0);
        const int dd = l & 15;
        const int b  = mt * 16 + M;
        atomicAdd(&s_raw[((size_t)b * N_OUT + o) * D_OUT + dd], sum);
    }
}

// ----------------------------------------------------------------- squash
// one thread per (b,o) row of 16; launch <<<10,256>>>
__global__ void squash_kernel(const float* __restrict__ s_raw,
                              float* __restrict__ out) {
    const int r = blockIdx.x * blockDim.x + threadIdx.x;   // 0..2559
    if (r >= BATCH * N_OUT) return;
    const float4* __restrict__ p = (const float4*)(s_raw + (size_t)r * D_OUT);
    float4 v0 = p[0], v1 = p[1], v2 = p[2], v3 = p[3];
    float n = v0.x * v0.x + v0.y * v0.y + v0.z * v0.z + v0.w * v0.w
            + v1.x * v1.x + v1.y * v1.y + v1.z * v1.z + v1.w * v1.w
            + v2.x * v2.x + v2.y * v2.y + v2.z * v2.z + v2.w * v2.w
            + v3.x * v3.x + v3.y * v3.y + v3.z * v3.z + v3.w * v3.w;
    // squash(v) = v * (|v| / (1 + |v|^2))
    const float scale = __fsqrt_rn(n) / (1.0f + n);
    float4* __restrict__ q = (float4*)(out + (size_t)r * D_OUT);
    v0.x *= scale; v0.y *= scale; v0.z *= scale; v0.w *= scale;
    v1.x *= scale; v1.y *= scale; v1.z *= scale; v1.w *= scale;
    v2.x *= scale; v2.y *= scale; v2.z *= scale; v2.w *= scale;
    v3.x *= scale; v3.y *= scale; v3.z *= scale; v3.w *= scale;
    q[0] = v0; q[1] = v1; q[2] = v2; q[3] = v3;
}

// ------------------------------------------------------- agreement update
// one block per input capsule i; x_i and W_i staged in LDS (async copy on
// CDNA5 when available); thread t owns batch row b=t; recomputes
// u_hat[b,i,o,:] in registers and dots with s.
__global__ void __launch_bounds__(256)
agree_kernel(const float* __restrict__ x, const float* __restrict__ W,
             const float* __restrict__ s, float* __restrict__ blog) {
    const int i = blockIdx.x;
    const int t = threadIdx.x;

    __shared__ float xs[BATCH * D_IN];                 // 8 KB
    __shared__ float wsl[N_OUT * D_OUT * D_IN];        // 5 KB
    __shared__ float accs[BATCH][N_OUT];               // 10 KB

#if USE_ASYNC_LDS
    // GLOBAL_LOAD_ASYNC_TO_LDS_B32: copy path writes LDS directly, no VGPR
    // round-trip; completion tracked with ASYNCcnt.
    for (int e = t; e < BATCH * D_IN; e += 256)
        __builtin_amdgcn_global_load_async_to_lds_b32(
            (g1int*)(x + (size_t)(e >> 3) * (N_IN * D_IN) + (size_t)i * D_IN + (e & 7)),
            (l3int*)&xs[e], 0, 0);
    for (int e = t; e < N_OUT * D_OUT * D_IN; e += 256)
        __builtin_amdgcn_global_load_async_to_lds_b32(
            (g1int*)(W + (size_t)i * (N_OUT * D_OUT * D_IN) + e),
            (l3int*)&wsl[e], 0, 0);
    __builtin_amdgcn_s_wait_asynccnt(0);   // s_wait_asynccnt 0
    __syncthreads();
#else
    for (int e = t; e < BATCH * D_IN; e += 256)
        xs[e] = x[(size_t)(e >> 3) * (N_IN * D_IN) + (size_t)i * D_IN + (e & 7)];
    for (int e = t; e < N_OUT * D_OUT * D_IN; e += 256)
        wsl[e] = W[(size_t)i * (N_OUT * D_OUT * D_IN) + e];
    __syncthreads();
#endif

    float xv[D_IN];
#pragma unroll
    for (int k = 0; k < D_IN; ++k) xv[k] = xs[t * D_IN + k];

#pragma unroll
    for (int o = 0; o < N_OUT; ++o) {
        const float4* __restrict__ sp =
            (const float4*)(s + ((size_t)t * N_OUT + o) * D_OUT);
        float4 q0 = sp[0], q1 = sp[1], q2 = sp[2], q3 = sp[3];
        float sv[D_OUT] = { q0.x, q0.y, q0.z, q0.w, q1.x, q1.y, q1.z, q1.w,
                            q2.x, q2.y, q2.z, q2.w, q3.x, q3.y, q3.z, q3.w };
        float a = 0.0f;
#pragma unroll
        for (int d = 0; d < D_OUT; ++d) {
            float uh = 0.0f;
#pragma unroll
            for (int k = 0; k < D_IN; ++k)
                uh = fmaf(wsl[(o * D_OUT + d) * D_IN + k], xv[k], uh);
            a = fmaf(uh, sv[d], a);
        }
        accs[t][o] = a;
    }
    __syncthreads();

    if (t < N_OUT) {
        float sum = 0.0f;
        for (int b = 0; b < BATCH; ++b) sum += accs[b][t];
        blog[(size_t)i * N_OUT + t] += sum;   // single writer per i -> no atomic
    }
}

// -------------------------------------------------------------- launcher
extern "C" void kernel_launch(void* const* d_in, const int* in_sizes, int n_in,
                              void* d_out, int out_size, void* d_ws, size_t ws_size,
                              hipStream_t stream) {
    const float* x = (const float*)d_in[0];   // [256,1152,8]
    const float* W = (const float*)d_in[1];   // [1152,10,16,8]
    float* out = (float*)d_out;               // [256,10,16]
    float* wsf = (float*)d_ws;

    float* blog = wsf;                        // 11520 f32
    float* c    = wsf + 11520;                // 11520 f32
    float* sraw = wsf + 23040;                // 40960 f32
    float* ssq  = wsf + 64000;                // 40960 f32  (total ~420 KB)

    zero_kernel<<<45, 256, 0, stream>>>(blog, N_IN * N_OUT);

    for (int r = 0; r < 4; ++r) {   // ROUTING_ITERS + 1
        softmax_zero_kernel<<<9, 128, 0, stream>>>(blog, c, sraw);
        s_gemm_kernel<<<dim3(16, N_OUT, KS), WAVES * 32, 0, stream>>>(x, W, c, sraw);
        squash_kernel<<<10, 256, 0, stream>>>(sraw, (r == 3) ? out : ssq);
        if (r < 3)
            agree_kernel<<<N_IN, 256, 0, stream>>>(x, W, ssq, blog);
    }
}